// FlattenedMultiStreamSystem_52321291600189
// MI455X (gfx1250) — compile-verified
//
#include <hip/hip_runtime.h>

// FlattenedMultiStreamSystem on MI455X (gfx1250).
// Bandwidth-bound scan: 256 steps x (complex 4096x4096 matvec). W (134 MB fp32)
// is L2-resident (192 MB L2), so each step streams W from L2. fp32 WMMA
// (V_WMMA_F32_16X16X4_F32) keeps reference precision; RHS vectors nLr/nLi are
// packed into B-matrix columns 0 and 1 so one accumulator carries Fr and Fi.

#define NDIM   4096
#define STEPS  256
#define TILE   16          // output rows per block (one WMMA M-tile)
#define TPB    256         // 8 wave32s per block
#define NWAVE  8
#define KCHUNK (NDIM / NWAVE)   // 512 K-columns per wave

typedef __attribute__((ext_vector_type(2))) float v2f;
typedef __attribute__((ext_vector_type(8))) float v8f;

// c = exp(Ar*dt)*cos(Ai*dt), s = exp(Ar*dt)*sin(Ai*dt), precomputed once.
__global__ void rot_precompute(const float* __restrict__ Ar,
                               const float* __restrict__ Ai,
                               const float* __restrict__ dtp,
                               float* __restrict__ c,
                               float* __restrict__ s) {
    int i = blockIdx.x * blockDim.x + threadIdx.x;
    if (i < NDIM) {
        float dt = dtp[0];
        float er = expf(Ar[i] * dt);
        float ph = Ai[i] * dt;
        c[i] = er * cosf(ph);
        s[i] = er * sinf(ph);
    }
}

__global__ __launch_bounds__(TPB, 1)
void step_kernel(const float* __restrict__ Lr_in,
                 const float* __restrict__ Li_in,
                 const float* __restrict__ rc,
                 const float* __restrict__ rs,
                 const float* __restrict__ Wr,
                 const float* __restrict__ Wi,
                 const float* __restrict__ war,
                 const float* __restrict__ wai,
                 const float* __restrict__ theta,
                 float* __restrict__ Lr_out,
                 float* __restrict__ Li_out) {
    __shared__ float nLr[NDIM];                    // rotated latent (real)
    __shared__ float nLi[NDIM];                    // rotated latent (imag)
    __shared__ float partial[NWAVE][2][TILE];      // per-wave Fr/Fi partials

    const int tid = threadIdx.x;

    // ---- Phase 1: rotate the full latent vector into LDS (redundant per
    // block; 64 KB of reads vs 512 KB of W reads -> negligible) ----
    for (int i = tid; i < NDIM; i += TPB) {
        float lr = Lr_in[i], li = Li_in[i];
        float cc = rc[i],    ss = rs[i];
        nLr[i] = lr * cc - li * ss;
        nLi[i] = lr * ss + li * cc;
    }
    __syncthreads();

    // ---- Phase 2: WMMA complex matvec over this block's 16-row tile ----
    const int wave  = tid >> 5;
    const int lane  = tid & 31;
    const int half  = lane >> 4;       // 0: K slots 0,1 ; 1: K slots 2,3
    const int col   = lane & 15;       // A-matrix row within tile / B column
    const int rbase = blockIdx.x * TILE;

    // B-column selectors: column 0 carries (nLr, -nLi), column 1 (nLi, nLr).
    const float f0 = (col == 0) ? 1.0f : 0.0f;
    const float f1 = (col == 1) ? 1.0f : 0.0f;

    const size_t rowOff = (size_t)(rbase + col) * NDIM;
    const float* wrp = Wr + rowOff;
    const float* wip = Wi + rowOff;

    v8f acc = {};
    const int kbase = wave * KCHUNK + 2 * half;    // 8-byte aligned
    #pragma unroll 4
    for (int k = 0; k < KCHUNK; k += 4) {
        const int kk = kbase + k;
        // A tiles: 16x4 f32 layout -> per-lane contiguous pair (b64 loads)
        v2f ar = *reinterpret_cast<const v2f*>(wrp + kk);
        v2f ai = *reinterpret_cast<const v2f*>(wip + kk);
        // RHS slabs from LDS (b64 loads)
        v2f xr = *reinterpret_cast<const v2f*>(&nLr[kk]);
        v2f xi = *reinterpret_cast<const v2f*>(&nLi[kk]);
        // Build B: col0 = nLr / -nLi, col1 = nLi / nLr, others zero.
        v2f br, bi;
        br.x = f0 * xr.x + f1 * xi.x;
        br.y = f0 * xr.y + f1 * xi.y;
        bi.x = f1 * xr.x - f0 * xi.x;
        bi.y = f1 * xr.y - f0 * xi.y;
        // C[:,0] += Wr*nLr - Wi*nLi  (Fr) ; C[:,1] += Wr*nLi + Wi*nLr (Fi)
        acc = __builtin_amdgcn_wmma_f32_16x16x4_f32(
                  false, ar, false, br, (short)0, acc, false, false);
        acc = __builtin_amdgcn_wmma_f32_16x16x4_f32(
                  false, ai, false, bi, (short)0, acc, false, false);
    }

    // ---- Phase 3: extract columns 0/1 of C.
    // Lane 0 holds (M=v, N=0) v=0..7; lane 16 holds (M=v+8, N=0);
    // lanes 1/17 the same for N=1. ----
    if (col <= 1) {
        const int rofs = half * 8;
        #pragma unroll
        for (int v = 0; v < 8; ++v)
            partial[wave][col][rofs + v] = acc[v];
    }
    __syncthreads();

    // ---- Phase 4: reduce 8 waves, apply collapse gate, write trajectory ----
    if (tid < TILE) {
        const int r = rbase + tid;
        float Fr = 0.0f, Fi = 0.0f;
        #pragma unroll
        for (int w = 0; w < NWAVE; ++w) {
            Fr += partial[w][0][tid];
            Fi += partial[w][1][tid];
        }
        const float lr = nLr[r], li = nLi[r];
        const float a = war[r] * lr + wai[r] * li;
        const bool  m = (a >= theta[r]);
        const float relFr = Fr > 0.0f ? Fr : 0.0f;
        Lr_out[r] = m ? lr * relFr : lr;
        Li_out[r] = m ? li * Fi    : li;
    }
}

extern "C" void kernel_launch(void* const* d_in, const int* in_sizes, int n_in,
                              void* d_out, int out_size, void* d_ws, size_t ws_size,
                              hipStream_t stream) {
    const float* Ar    = (const float*)d_in[0];
    const float* Ai    = (const float*)d_in[1];
    const float* war   = (const float*)d_in[2];
    const float* wai   = (const float*)d_in[3];
    const float* theta = (const float*)d_in[4];
    const float* Wr    = (const float*)d_in[5];
    const float* Wi    = (const float*)d_in[6];
    const float* Lr0   = (const float*)d_in[7];
    const float* Li0   = (const float*)d_in[8];
    const float* dt    = (const float*)d_in[9];

    float* rc = (float*)d_ws;          // 4096 floats
    float* rs = rc + NDIM;             // 4096 floats (32 KB total scratch)

    float* outLr = (float*)d_out;                       // (STEPS, N)
    float* outLi = outLr + (size_t)STEPS * NDIM;        // (STEPS, N)

    rot_precompute<<<(NDIM + 255) / 256, 256, 0, stream>>>(Ar, Ai, dt, rc, rs);

    const float* pLr = Lr0;
    const float* pLi = Li0;
    for (int t = 0; t < STEPS; ++t) {
        float* oLr = outLr + (size_t)t * NDIM;
        float* oLi = outLi + (size_t)t * NDIM;
        step_kernel<<<NDIM / TILE, TPB, 0, stream>>>(
            pLr, pLi, rc, rs, Wr, Wi, war, wai, theta, oLr, oLi);
        pLr = oLr;   // next step consumes this step's trajectory row
        pLi = oLi;
    }
}